// DigitalCapsule_9234179687266
// MI455X (gfx1250) — compile-verified
//
#include <hip/hip_runtime.h>
#include <math.h>

typedef __attribute__((ext_vector_type(2))) float v2f;
typedef __attribute__((ext_vector_type(8))) float v8f;

#define B_   64
#define N_   1152
#define O_   32
#define DO_  16
#define DI_  8
#define OD_  (O_ * DO_)   // 512
#define EPS_ 1e-7f

// ---------------------------------------------------------------------------
// Kernel 1: x_hat[b,n,(o,i)] = sum_j weight[n,(o,i),j] * x[b,n,j]
// Batched GEMM per n: [64 x 8] x [8 x 512] using V_WMMA_F32_16X16X4_F32,
// two K-steps (K=8 = 2 x 4), fp32 accumulate.
// grid = (1152, 8), block = 128 (4 waves). Wave w -> M-tile (rows 16w..16w+15),
// blockIdx.y -> 64-column group (4 column tiles of 16).
// ---------------------------------------------------------------------------
__global__ __launch_bounds__(128) void k_xhat(const float* __restrict__ x,
                                              const float* __restrict__ w,
                                              float* __restrict__ xhat)
{
    const int n    = blockIdx.x;
    const int cg   = blockIdx.y;
    const int wv   = threadIdx.x >> 5;   // wave in workgroup: M tile
    const int lane = threadIdx.x & 31;
    const int lm   = lane & 15;
    const int kh   = (lane >> 4) << 1;   // 0 (lanes 0-15) or 2 (lanes 16-31)
    const int b0   = wv * 16;

    // A tile: 16x4 f32 layout -> lane lm holds K = kh, kh+1 in v0,v1.
    // Step0 covers K=0..3, step1 covers K=4..7.
    const float* xp = x + ((size_t)(b0 + lm) * N_ + n) * DI_;
    v2f a_lo, a_hi;
    a_lo[0] = xp[kh];     a_lo[1] = xp[kh + 1];
    a_hi[0] = xp[kh + 4]; a_hi[1] = xp[kh + 5];

#pragma unroll
    for (int ct = 0; ct < 4; ++ct) {
        const int col0 = cg * 64 + ct * 16;
        // B tile: 4x16 f32; column (o,i) has Di contiguous in memory.
        const float* wp = w + ((size_t)n * OD_ + col0 + lm) * DI_;
        v2f b_lo, b_hi;
        b_lo[0] = wp[kh];     b_lo[1] = wp[kh + 1];
        b_hi[0] = wp[kh + 4]; b_hi[1] = wp[kh + 5];

        v8f acc = {};
        acc = __builtin_amdgcn_wmma_f32_16x16x4_f32(false, a_lo, false, b_lo,
                                                    (short)0, acc, false, false);
        acc = __builtin_amdgcn_wmma_f32_16x16x4_f32(false, a_hi, false, b_hi,
                                                    (short)0, acc, false, false);

        // D layout: VGPR v -> M = v (lanes 0-15) or 8+v (lanes 16-31), N = lm.
        const int bhi = (lane >> 4) * 8;
#pragma unroll
        for (int v = 0; v < 8; ++v) {
            const int brow = b0 + bhi + v;
            xhat[((size_t)brow * N_ + n) * OD_ + col0 + lm] = acc[v];
        }
    }
}

// ---------------------------------------------------------------------------
// Kernel 2: s[b,o,:] = sum_n c[b,n,o] * x_hat[b,n,o,:]; then squash -> vj.
// grid = (64, 32), block = 256. Each thread strides n; LDS tree reduction.
// iter 0: b==0 => c uniform 1/32 (uniform_c flag).
// ---------------------------------------------------------------------------
__global__ __launch_bounds__(256) void k_s_squash(const float* __restrict__ xhat,
                                                  const float* __restrict__ c,
                                                  float* __restrict__ vj_out,
                                                  int uniform_c)
{
    const int b   = blockIdx.x;
    const int o   = blockIdx.y;
    const int tid = threadIdx.x;

    float acc[DO_];
#pragma unroll
    for (int i = 0; i < DO_; ++i) acc[i] = 0.f;

    for (int n = tid; n < N_; n += 256) {
        const float cv = uniform_c ? (1.0f / (float)O_)
                                   : c[((size_t)b * N_ + n) * O_ + o];
        const float* xp = xhat + ((size_t)b * N_ + n) * OD_ + o * DO_;
#pragma unroll
        for (int q = 0; q < 4; ++q) {
            const float4 v = *(const float4*)(xp + 4 * q);
            acc[4 * q + 0] += cv * v.x;
            acc[4 * q + 1] += cv * v.y;
            acc[4 * q + 2] += cv * v.z;
            acc[4 * q + 3] += cv * v.w;
        }
    }

    __shared__ float red[DO_ * 256];
#pragma unroll
    for (int i = 0; i < DO_; ++i) red[i * 256 + tid] = acc[i];
    __syncthreads();

    for (int stride = 128; stride > 0; stride >>= 1) {
        if (tid < stride) {
#pragma unroll
            for (int i = 0; i < DO_; ++i)
                red[i * 256 + tid] += red[i * 256 + tid + stride];
        }
        __syncthreads();
    }

    if (tid == 0) {
        float n2 = 0.f;
#pragma unroll
        for (int i = 0; i < DO_; ++i) {
            const float sv = red[i * 256];
            n2 += sv * sv;
        }
        const float scale = (n2 / (1.0f + n2)) / sqrtf(n2 + EPS_);
        float* vo = vj_out + ((size_t)b * O_ + o) * DO_;
#pragma unroll
        for (int i = 0; i < DO_; ++i) vo[i] = scale * red[i * 256];
    }
}

// ---------------------------------------------------------------------------
// Kernel 3 (fused agreement + softmax): one wave per (b,n), lane = o.
//   b_new = b_old + <x_hat[b,n,o,:], vj[b,o,:]> ; c = softmax_o(b_new)
// wave32 shuffle reductions for max/sum across the 32 output capsules.
// grid = 9216, block = 256 (8 waves) -> 73728 waves = B*N.
// ---------------------------------------------------------------------------
__global__ __launch_bounds__(256) void k_agree_softmax(const float* __restrict__ xhat,
                                                       const float* __restrict__ vj,
                                                       const float* __restrict__ b_in,
                                                       float* __restrict__ b_out,
                                                       float* __restrict__ c_out,
                                                       int b_is_zero)
{
    const int gtid = blockIdx.x * blockDim.x + threadIdx.x;
    const int wid  = gtid >> 5;          // 0 .. B*N-1
    const int lane = gtid & 31;          // o
    const int b    = wid / N_;
    const int n    = wid % N_;

    const float* xp = xhat + ((size_t)b * N_ + n) * OD_ + lane * DO_;
    const float* vp = vj + ((size_t)b * O_ + lane) * DO_;

    float dot = 0.f;
#pragma unroll
    for (int q = 0; q < 4; ++q) {
        const float4 xv = *(const float4*)(xp + 4 * q);
        const float4 vv = *(const float4*)(vp + 4 * q);
        dot += xv.x * vv.x + xv.y * vv.y + xv.z * vv.z + xv.w * vv.w;
    }

    const size_t idx = ((size_t)b * N_ + n) * O_ + lane;
    const float bnew = (b_is_zero ? 0.f : b_in[idx]) + dot;
    b_out[idx] = bnew;

    // softmax over the 32 lanes (output capsules) of this wave
    float m = bnew;
#pragma unroll
    for (int msk = 1; msk < 32; msk <<= 1)
        m = fmaxf(m, __shfl_xor(m, msk, 32));
    const float e = expf(bnew - m);
    float s = e;
#pragma unroll
    for (int msk = 1; msk < 32; msk <<= 1)
        s += __shfl_xor(s, msk, 32);

    c_out[idx] = e / s;
}

// ---------------------------------------------------------------------------
// Host-side launch. Workspace layout (floats):
//   xhat : 64*1152*512  = 37,748,736   (151 MB -> resident in 192 MB L2)
//   b    : 64*1152*32   =  2,359,296
//   c    : 64*1152*32   =  2,359,296
//   vj   : 64*32*16     =     32,768
// ---------------------------------------------------------------------------
extern "C" void kernel_launch(void* const* d_in, const int* in_sizes, int n_in,
                              void* d_out, int out_size, void* d_ws, size_t ws_size,
                              hipStream_t stream) {
    (void)in_sizes; (void)n_in; (void)out_size; (void)ws_size;

    const float* x = (const float*)d_in[0];
    const float* w = (const float*)d_in[1];
    float* out = (float*)d_out;

    float* xhat = (float*)d_ws;
    float* bws  = xhat + (size_t)B_ * N_ * OD_;
    float* cws  = bws + (size_t)B_ * N_ * O_;
    float* vjws = cws + (size_t)B_ * N_ * O_;

    // Stage 1: x_hat via fp32 WMMA
    k_xhat<<<dim3(N_, 8), 128, 0, stream>>>(x, w, xhat);

    // Routing iteration 0: c uniform (b == 0)
    k_s_squash<<<dim3(B_, O_), 256, 0, stream>>>(xhat, nullptr, vjws, 1);
    k_agree_softmax<<<(B_ * N_ * 32) / 256, 256, 0, stream>>>(xhat, vjws, nullptr,
                                                              bws, cws, 1);
    // Routing iteration 1
    k_s_squash<<<dim3(B_, O_), 256, 0, stream>>>(xhat, cws, vjws, 0);
    k_agree_softmax<<<(B_ * N_ * 32) / 256, 256, 0, stream>>>(xhat, vjws, bws,
                                                              bws, cws, 0);
    // Routing iteration 2 (final): write vj straight to d_out
    k_s_squash<<<dim3(B_, O_), 256, 0, stream>>>(xhat, cws, out, 0);
}